// MTModel_28690381537878
// MI455X (gfx1250) — compile-verified
//
#include <hip/hip_runtime.h>
#include <hip/hip_bf16.h>
#include <math.h>

#define D   768
#define D2  1536
#define D3  2304
#define D6  4608
#define BSZ 16
#define SEQ 256
#define SM1 255
#define LOUT 7
#define NSTEP 6

typedef __attribute__((ext_vector_type(2))) float v2f;
typedef __attribute__((ext_vector_type(8))) float v8f;

struct P {
    const float* hs[3];       // text, audio, video hidden states (13,B,S,D)
    const float* init[3];     // (B,D)
    const float* lnc_g; const float* lnc_b;
    const float* ln2_g; const float* ln2_b;
    const float* lnm_g[3]; const float* lnm_b[3];   // lnt, lna, lnv
    const float* lnf_g; const float* lnf_b;
    const float* Wlc; const float* blc;
    const float* W1[3];  const float* b1[3];
    const float* Watt[3]; const float* batt[3];
    const float* W2a[3]; const float* b2a[3];
    const float* W2b[3]; const float* b2b[3];
    const float* W2c[3]; const float* b2c[3];
    const float* lnp_g[3]; const float* lnp_b[3];
    const float* Wp[3]; const float* bp[3];
    const float* Wf; const float* bf; const float* wf;
    // workspace
    float* curr;    // [3][B][D]
    float* currln;  // [3][B][D]
    float* c1;      // [3][B][D]
    float* xc;      // [3][B][2D]
    float* mp;      // [3][B][D]   (mp_pre then mp)
    float* cat3;    // [3][B][3D]
    float* m2;      // [3][B][D]
    float* out;     // d_out
};

// ---------------- reductions ----------------
__device__ __forceinline__ float waveReduceSum(float v) {
    #pragma unroll
    for (int off = 16; off > 0; off >>= 1) v += __shfl_xor(v, off, 32);
    return v;
}
__device__ __forceinline__ float waveReduceMax(float v) {
    #pragma unroll
    for (int off = 16; off > 0; off >>= 1) v = fmaxf(v, __shfl_xor(v, off, 32));
    return v;
}
__device__ __forceinline__ float blockReduceSum(float v, float* scratch) {
    int lane = threadIdx.x & 31, wv = threadIdx.x >> 5;
    v = waveReduceSum(v);
    if (lane == 0) scratch[wv] = v;
    __syncthreads();
    int nw = blockDim.x >> 5;
    float t = (threadIdx.x < (unsigned)nw) ? scratch[threadIdx.x] : 0.0f;
    if (wv == 0) {
        t = waveReduceSum(t);
        if (lane == 0) scratch[0] = t;
    }
    __syncthreads();
    float r = scratch[0];
    __syncthreads();
    return r;
}
__device__ __forceinline__ float blockReduceMax(float v, float* scratch) {
    int lane = threadIdx.x & 31, wv = threadIdx.x >> 5;
    v = waveReduceMax(v);
    if (lane == 0) scratch[wv] = v;
    __syncthreads();
    int nw = blockDim.x >> 5;
    float t = (threadIdx.x < (unsigned)nw) ? scratch[threadIdx.x] : -1e30f;
    if (wv == 0) {
        t = waveReduceMax(t);
        if (lane == 0) scratch[0] = t;
    }
    __syncthreads();
    float r = scratch[0];
    __syncthreads();
    return r;
}
// whole-block layernorm of one row of length n
__device__ __forceinline__ void ln_row(const float* x, const float* g, const float* bb,
                                       float* out, int n, float* scratch) {
    float s = 0.f, s2 = 0.f;
    for (int i = threadIdx.x; i < n; i += blockDim.x) { float v = x[i]; s += v; s2 += v * v; }
    float tot  = blockReduceSum(s,  scratch);
    float tot2 = blockReduceSum(s2, scratch);
    float mean = tot / (float)n;
    float var  = tot2 / (float)n - mean * mean;
    float rstd = rsqrtf(var + 1e-5f);
    for (int i = threadIdx.x; i < n; i += blockDim.x)
        out[i] = (x[i] - mean) * rstd * g[i] + bb[i];
}

// ---------------- per-step kernels ----------------
// LNs of curr + three cls rows; builds xc = [ln(c2), ln(c3)]
__global__ void k_prep(P p, int step, int first) {
    int b = blockIdx.x, m = blockIdx.y;
    __shared__ float scratch[32];
    const int h2 = (m == 0) ? 1 : 0;
    const int h3 = (m == 2) ? 1 : 2;
    const float* csrc = first ? (p.init[m] + (size_t)b * D)
                              : (p.curr + ((size_t)m * BSZ + b) * D);
    ln_row(csrc, p.lnc_g, p.lnc_b, p.currln + ((size_t)m * BSZ + b) * D, D, scratch);
    const float* cls1 = p.hs[m]  + (((size_t)step * BSZ + b) * SEQ) * D;
    ln_row(cls1, p.lnm_g[m], p.lnm_b[m], p.c1 + ((size_t)m * BSZ + b) * D, D, scratch);
    const float* cls2 = p.hs[h2] + (((size_t)step * BSZ + b) * SEQ) * D;
    ln_row(cls2, p.lnm_g[h2], p.lnm_b[h2], p.xc + ((size_t)m * BSZ + b) * D2, D, scratch);
    const float* cls3 = p.hs[h3] + (((size_t)step * BSZ + b) * SEQ) * D;
    ln_row(cls3, p.lnm_g[h3], p.lnm_b[h3], p.xc + ((size_t)m * BSZ + b) * D2 + D, D, scratch);
}

// mp_pre = xc @ W1 + b1   (M=16, N=768, K=1536) via v_wmma_f32_16x16x4_f32
__global__ void k_gemm_w1(P p) {
    int m = blockIdx.y;
    int wave = threadIdx.x >> 5, lane = threadIdx.x & 31;
    int n0 = (blockIdx.x * 8 + wave) * 16;
    int rn = lane & 15, hi = lane >> 4;    // rn: A row / B col index; hi: K-half
    int col = n0 + rn;
    const float* Arow = p.xc + ((size_t)m * BSZ + rn) * D2;
    const float* W = p.W1[m];
    v8f acc = {};
    for (int k0 = 0; k0 < D2; k0 += 4) {
        int ka = k0 + 2 * hi;
        v2f a; a.x = Arow[ka]; a.y = Arow[ka + 1];
        v2f bv; bv.x = W[(size_t)ka * D + col]; bv.y = W[(size_t)(ka + 1) * D + col];
        acc = __builtin_amdgcn_wmma_f32_16x16x4_f32(false, a, false, bv, (short)0, acc,
                                                    false, false);
    }
    float bc = p.b1[m][col];          // hoisted: single load, no alias with stores below
    float res[8];
    #pragma unroll
    for (int r = 0; r < 8; ++r) res[r] = acc[r] + bc;
    float* out = p.mp + (size_t)m * BSZ * D;
    #pragma unroll
    for (int r = 0; r < 8; ++r) {
        int mr = r + 8 * hi;
        out[(size_t)mr * D + col] = res[r];
    }
}

// mp = ln(mp_pre); cat3 = [mp, currln, c1]
__global__ void k_mid(P p) {
    int b = blockIdx.x, m = blockIdx.y;
    __shared__ float scratch[32];
    float* mprow = p.mp + ((size_t)m * BSZ + b) * D;
    ln_row(mprow, p.lnc_g, p.lnc_b, mprow, D, scratch);
    __syncthreads();
    float* cat = p.cat3 + ((size_t)m * BSZ + b) * D3;
    const float* cl = p.currln + ((size_t)m * BSZ + b) * D;
    const float* c1 = p.c1 + ((size_t)m * BSZ + b) * D;
    for (int i = threadIdx.x; i < D; i += blockDim.x) {
        cat[i] = mprow[i]; cat[D + i] = cl[i]; cat[D2 + i] = c1[i];
    }
}

// three gate GEMMs sharing A loads (K=2304), fused sigmoid gating -> m2
__global__ void k_gemm_w2(P p) {
    int m = blockIdx.y;
    int wave = threadIdx.x >> 5, lane = threadIdx.x & 31;
    int n0 = (blockIdx.x * 8 + wave) * 16;
    int rn = lane & 15, hi = lane >> 4;
    int col = n0 + rn;
    const float* Arow = p.cat3 + ((size_t)m * BSZ + rn) * D3;
    const float* WA = p.W2a[m];
    const float* WB = p.W2b[m];
    const float* WC = p.W2c[m];
    v8f aa = {}, ab = {}, ac = {};
    for (int k0 = 0; k0 < D3; k0 += 4) {
        int ka = k0 + 2 * hi;
        v2f a; a.x = Arow[ka]; a.y = Arow[ka + 1];
        v2f b0; b0.x = WA[(size_t)ka * D + col]; b0.y = WA[(size_t)(ka + 1) * D + col];
        aa = __builtin_amdgcn_wmma_f32_16x16x4_f32(false, a, false, b0, (short)0, aa, false, false);
        v2f b1v; b1v.x = WB[(size_t)ka * D + col]; b1v.y = WB[(size_t)(ka + 1) * D + col];
        ab = __builtin_amdgcn_wmma_f32_16x16x4_f32(false, a, false, b1v, (short)0, ab, false, false);
        v2f b2v; b2v.x = WC[(size_t)ka * D + col]; b2v.y = WC[(size_t)(ka + 1) * D + col];
        ac = __builtin_amdgcn_wmma_f32_16x16x4_f32(false, a, false, b2v, (short)0, ac, false, false);
    }
    // hoist all epilogue loads into registers BEFORE any store (avoid alias ordering)
    float bav = p.b2a[m][col], bbv = p.b2b[m][col], bcv = p.b2c[m][col];
    const float* mp = p.mp     + (size_t)m * BSZ * D;
    const float* cl = p.currln + (size_t)m * BSZ * D;
    const float* c1 = p.c1     + (size_t)m * BSZ * D;
    float res[8];
    #pragma unroll
    for (int r = 0; r < 8; ++r) {
        int mr = r + 8 * hi;
        size_t idx = (size_t)mr * D + col;
        float ga = 1.f / (1.f + __expf(-(aa[r] + bav)));
        float gb = 1.f / (1.f + __expf(-(ab[r] + bbv)));
        float gc = 1.f / (1.f + __expf(-(ac[r] + bcv)));
        res[r] = ga * mp[idx] + gb * cl[idx] + gc * c1[idx];
    }
    float* m2 = p.m2 + (size_t)m * BSZ * D;
    #pragma unroll
    for (int r = 0; r < 8; ++r) {
        int mr = r + 8 * hi;
        m2[(size_t)mr * D + col] = res[r];
    }
}

// fused LN+score, softmax, weighted sum, ln2, per-step preds. One block per (b,m).
__global__ void k_attn(P p, int step, int sidx) {
    int b = blockIdx.x, m = blockIdx.y;
    int tid = threadIdx.x, lane = tid & 31, wave = tid >> 5;
    __shared__ float scratch[32];
    __shared__ __align__(16) float sGW[D];
    __shared__ float sScore[SM1], sMean[SM1], sRstd[SM1];
    __shared__ float sY[D];
    __shared__ float sBias, sGw, sBw;
    const float* g1 = p.lnm_g[m]; const float* b1 = p.lnm_b[m];
    const float* wa = p.Watt[m];
    const float* m2row = p.m2 + ((size_t)m * BSZ + b) * D;
    float gw = 0.f, bw = 0.f, bi = 0.f;
    for (int d = tid; d < D; d += blockDim.x) {
        float w0 = wa[d];
        sGW[d] = g1[d] * w0;
        gw += g1[d] * w0;
        bw += b1[d] * w0;
        bi += m2row[d] * wa[D + d];
    }
    gw = blockReduceSum(gw, scratch);
    bw = blockReduceSum(bw, scratch);
    bi = blockReduceSum(bi, scratch);
    if (tid == 0) { sBias = bi + p.batt[m][0]; sGw = gw; sBw = bw; }
    __syncthreads();
    const float* hbase = p.hs[m] + (((size_t)step * BSZ + b) * SEQ) * D;
    // pass 1 (bandwidth-critical): float4 row loads; fused LN+dot
    // score = rstd*(sum(x*g*Wa) - mean*sum(g*Wa)) + sum(b*Wa) + bias
    for (int s = wave; s < SM1; s += 8) {
        const float4* x4 = (const float4*)(hbase + (size_t)(s + 1) * D);
        if (s + 8 < SM1) __builtin_prefetch(hbase + (size_t)(s + 9) * D + lane * 4, 0, 1);
        float su = 0.f, sq = 0.f, sw = 0.f;
        for (int q = lane; q < (D / 4); q += 32) {
            float4 v = x4[q];
            float4 g4 = *(const float4*)&sGW[q * 4];
            su += (v.x + v.y) + (v.z + v.w);
            sq += v.x * v.x + v.y * v.y + v.z * v.z + v.w * v.w;
            sw += v.x * g4.x + v.y * g4.y + v.z * g4.z + v.w * g4.w;
        }
        su = waveReduceSum(su); sq = waveReduceSum(sq); sw = waveReduceSum(sw);
        if (lane == 0) {
            float mean = su * (1.0f / D);
            float var  = sq * (1.0f / D) - mean * mean;
            float rstd = rsqrtf(var + 1e-5f);
            sMean[s] = mean; sRstd[s] = rstd;
            sScore[s] = (sw - mean * sGw) * rstd + sBw + sBias;
        }
    }
    __syncthreads();
    // softmax over 255
    float mx = -1e30f;
    for (int s = tid; s < SM1; s += blockDim.x) mx = fmaxf(mx, sScore[s]);
    mx = blockReduceMax(mx, scratch);
    float se = 0.f;
    for (int s = tid; s < SM1; s += blockDim.x) { float e = __expf(sScore[s] - mx); sScore[s] = e; se += e; }
    se = blockReduceSum(se, scratch);
    float inv = 1.0f / se;
    for (int s = tid; s < SM1; s += blockDim.x) sScore[s] *= inv;
    __syncthreads();
    // pass 2 (L2-resident re-read): new_t[d] = sum_s attn[s]*ln(x[s,d])
    int d0 = tid, d1 = tid + 256, d2 = tid + 512;
    float g0 = g1[d0], gA = g1[d1], gB = g1[d2];
    float h0 = b1[d0], hA = b1[d1], hB = b1[d2];
    float a0 = 0.f, a1 = 0.f, a2 = 0.f;
    for (int s = 0; s < SM1; ++s) {
        const float* x = hbase + (size_t)(s + 1) * D;
        float at = sScore[s], mean = sMean[s], rstd = sRstd[s];
        a0 += at * ((x[d0] - mean) * rstd * g0 + h0);
        a1 += at * ((x[d1] - mean) * rstd * gA + hA);
        a2 += at * ((x[d2] - mean) * rstd * gB + hB);
    }
    sY[d0] = a0 + m2row[d0]; sY[d1] = a1 + m2row[d1]; sY[d2] = a2 + m2row[d2];
    __syncthreads();
    // curr = ln2(new_t + m2)
    float su = 0.f, sq = 0.f;
    for (int d = tid; d < D; d += blockDim.x) { float v = sY[d]; su += v; sq += v * v; }
    su = blockReduceSum(su, scratch);
    sq = blockReduceSum(sq, scratch);
    float mean = su * (1.0f / D);
    float var  = sq * (1.0f / D) - mean * mean;
    float rstd = rsqrtf(var + 1e-5f);
    float* currow = p.curr + ((size_t)m * BSZ + b) * D;
    for (int d = tid; d < D; d += blockDim.x) {
        float cv = (sY[d] - mean) * rstd * p.ln2_g[d] + p.ln2_b[d];
        currow[d] = cv;
        sY[d] = cv;
    }
    __syncthreads();
    // preds[step] = curr @ Wlc + blc  -> out[112 + m*6*B*L + (sidx*B+b)*L + l]
    if (wave < LOUT) {
        float s7 = 0.f;
        for (int d = lane; d < D; d += 32) s7 += sY[d] * p.Wlc[(size_t)d * LOUT + wave];
        s7 = waveReduceSum(s7);
        if (lane == 0) {
            size_t off = (size_t)BSZ * LOUT + (size_t)m * NSTEP * BSZ * LOUT
                         + ((size_t)sidx * BSZ + b) * LOUT + wave;
            p.out[off] = s7 + p.blc[wave];
        }
    }
}

// final fusion: tp/vp/ap projections + fused 6D LN/GEMM + weighted combine
__global__ void k_final(P p) {
    int b = blockIdx.x, tid = threadIdx.x, lane = tid & 31, wave = tid >> 5;
    __shared__ float scratch[32];
    __shared__ float sF[D6];
    __shared__ float sT[D];
    __shared__ float sP[3 * LOUT];
    const float* seg0 = p.curr + ((size_t)0 * BSZ + b) * D;
    const float* seg1 = p.hs[0] + (((size_t)12 * BSZ + b) * SEQ) * D;
    const float* seg2 = p.curr + ((size_t)2 * BSZ + b) * D;
    const float* seg3 = p.hs[2] + (((size_t)12 * BSZ + b) * SEQ) * D;
    const float* seg4 = p.curr + ((size_t)1 * BSZ + b) * D;
    const float* seg5 = p.hs[1] + (((size_t)12 * BSZ + b) * SEQ) * D;
    for (int i = tid; i < D6; i += blockDim.x) {
        int sgi = i / D, off = i % D;
        const float* src = (sgi == 0) ? seg0 : (sgi == 1) ? seg1 : (sgi == 2) ? seg2
                         : (sgi == 3) ? seg3 : (sgi == 4) ? seg4 : seg5;
        sF[i] = src[off];
    }
    for (int mm = 0; mm < 3; ++mm) {
        const float* cls = p.hs[mm] + (((size_t)12 * BSZ + b) * SEQ) * D;
        __syncthreads();
        ln_row(cls, p.lnp_g[mm], p.lnp_b[mm], sT, D, scratch);
        __syncthreads();
        if (wave < LOUT) {
            float s = 0.f;
            for (int d = lane; d < D; d += 32) s += sT[d] * p.Wp[mm][(size_t)d * LOUT + wave];
            s = waveReduceSum(s);
            if (lane == 0) sP[mm * LOUT + wave] = s + p.bp[mm][wave];
        }
    }
    __syncthreads();
    ln_row(sF, p.lnf_g, p.lnf_b, sF, D6, scratch);
    __syncthreads();
    if (wave < LOUT) {
        float s = 0.f;
        for (int i = lane; i < D6; i += 32) s += sF[i] * p.Wf[(size_t)i * LOUT + wave];
        s = waveReduceSum(s);
        if (lane == 0) {
            float mmv = s + p.bf[wave];
            float r = sP[0 * LOUT + wave] * p.wf[0]   // tp
                    + sP[2 * LOUT + wave] * p.wf[1]   // vp
                    + sP[1 * LOUT + wave] * p.wf[2]   // ap
                    + mmv * p.wf[3];
            p.out[(size_t)b * LOUT + wave] = r;
        }
    }
}

extern "C" void kernel_launch(void* const* d_in, const int* in_sizes, int n_in,
                              void* d_out, int out_size, void* d_ws, size_t ws_size,
                              hipStream_t stream) {
    (void)in_sizes; (void)n_in; (void)out_size; (void)ws_size;
    P p;
    p.hs[0] = (const float*)d_in[0];
    p.hs[1] = (const float*)d_in[1];
    p.hs[2] = (const float*)d_in[2];
    p.init[0] = (const float*)d_in[3];
    p.init[1] = (const float*)d_in[4];
    p.init[2] = (const float*)d_in[5];
    p.lnc_g = (const float*)d_in[6];  p.lnc_b = (const float*)d_in[7];
    p.ln2_g = (const float*)d_in[8];  p.ln2_b = (const float*)d_in[9];
    p.lnm_g[0] = (const float*)d_in[10]; p.lnm_b[0] = (const float*)d_in[11]; // lnt
    p.lnm_g[1] = (const float*)d_in[12]; p.lnm_b[1] = (const float*)d_in[13]; // lna
    p.lnm_g[2] = (const float*)d_in[14]; p.lnm_b[2] = (const float*)d_in[15]; // lnv
    p.lnf_g = (const float*)d_in[16]; p.lnf_b = (const float*)d_in[17];
    p.Wlc = (const float*)d_in[18];   p.blc = (const float*)d_in[19];
    for (int m = 0; m < 3; ++m) {
        int base = 20 + m * 14;
        p.W1[m]    = (const float*)d_in[base + 0];
        p.b1[m]    = (const float*)d_in[base + 1];
        p.Watt[m]  = (const float*)d_in[base + 2];
        p.batt[m]  = (const float*)d_in[base + 3];
        p.W2a[m]   = (const float*)d_in[base + 4];
        p.b2a[m]   = (const float*)d_in[base + 5];
        p.W2b[m]   = (const float*)d_in[base + 6];
        p.b2b[m]   = (const float*)d_in[base + 7];
        p.W2c[m]   = (const float*)d_in[base + 8];
        p.b2c[m]   = (const float*)d_in[base + 9];
        p.lnp_g[m] = (const float*)d_in[base + 10];
        p.lnp_b[m] = (const float*)d_in[base + 11];
        p.Wp[m]    = (const float*)d_in[base + 12];
        p.bp[m]    = (const float*)d_in[base + 13];
    }
    p.Wf = (const float*)d_in[62];
    p.bf = (const float*)d_in[63];
    p.wf = (const float*)d_in[64];

    float* w = (float*)d_ws;
    p.curr   = w; w += (size_t)3 * BSZ * D;
    p.currln = w; w += (size_t)3 * BSZ * D;
    p.c1     = w; w += (size_t)3 * BSZ * D;
    p.xc     = w; w += (size_t)3 * BSZ * D2;
    p.mp     = w; w += (size_t)3 * BSZ * D;
    p.cat3   = w; w += (size_t)3 * BSZ * D3;
    p.m2     = w; w += (size_t)3 * BSZ * D;
    p.out = (float*)d_out;

    for (int si = 0; si < NSTEP; ++si) {
        int step = 6 + si;
        k_prep   <<<dim3(BSZ, 3), 256, 0, stream>>>(p, step, si == 0 ? 1 : 0);
        k_gemm_w1<<<dim3(6, 3),   256, 0, stream>>>(p);
        k_mid    <<<dim3(BSZ, 3), 256, 0, stream>>>(p);
        k_gemm_w2<<<dim3(6, 3),   256, 0, stream>>>(p);
        k_attn   <<<dim3(BSZ, 3), 256, 0, stream>>>(p, step, si);
    }
    k_final<<<BSZ, 256, 0, stream>>>(p);
}